// PhysicsLoss_56899726737553
// MI455X (gfx1250) — compile-verified
//
#include <hip/hip_runtime.h>
#include <hip/hip_bf16.h>

typedef __attribute__((ext_vector_type(2))) float v2f;
typedef __attribute__((ext_vector_type(8))) float v8f;

#define HH 128
#define WW 128
#define HWSZ (HH * WW)
#define CHW (3 * HWSZ)
#define BATCH 256
#define NINT 126                      // interior size per axis
#define NTOT_INT (BATCH * NINT * NINT)
#define NBC_PER_B 988
#define NTOT_BC (BATCH * NBC_PER_B)

#define NB_BASE 1024
#define NB_INT  1024
#define NB_BC   64

// ---------------- reduction helpers ----------------

__device__ __forceinline__ float waveSum32(float t) {
    #pragma unroll
    for (int off = 16; off > 0; off >>= 1) t += __shfl_xor(t, off, 32);
    return t;
}

// ---------------- kernel 1: base MSE ----------------

__global__ void k_base(const float4* __restrict__ o, const float4* __restrict__ t,
                       float* __restrict__ part, int n4) {
    float s = 0.f;
    for (int i = blockIdx.x * blockDim.x + threadIdx.x; i < n4;
         i += gridDim.x * blockDim.x) {
        float4 a = o[i], b = t[i];
        float dx = a.x - b.x, dy = a.y - b.y, dz = a.z - b.z, dw = a.w - b.w;
        s += dx * dx + dy * dy + dz * dz + dw * dw;
    }
    __shared__ float lds[8];
    int lane = threadIdx.x & 31, wid = threadIdx.x >> 5;
    s = waveSum32(s);
    if (lane == 0) lds[wid] = s;
    __syncthreads();
    if (threadIdx.x == 0) {
        float a = 0.f;
        #pragma unroll
        for (int w = 0; w < 8; ++w) a += lds[w];
        part[blockIdx.x] = a;
    }
}

// ---------------- kernel 2: boundary MSE ----------------

__global__ void k_bc(const float* __restrict__ outp, float* __restrict__ part) {
    float s = 0.f;
    for (int i = blockIdx.x * blockDim.x + threadIdx.x; i < NTOT_BC;
         i += gridDim.x * blockDim.x) {
        int b = i / NBC_PER_B;
        int r = i - b * NBC_PER_B;
        const float* ob = outp + b * CHW;
        float val;
        if (r < 18) {                      // out[:,1,55:73,0] - 0.01
            val = ob[HWSZ + (55 + r) * WW + 0] - 0.01f;
        } else if (r < 36) {               // out[:,0,55:73,127]
            val = ob[(55 + (r - 18)) * WW + 127];
        } else {
            int r2 = r - 36;
            int g  = r2 / 238;
            int t  = r2 - g * 238;
            int ch   = 1 + (g >> 1);       // groups 0,1 -> ch1 ; 2,3 -> ch2
            int edge = (g & 1) ? 127 : 0;  // idx 0 / -1
            int row, col;
            if (t < 128)      { row = edge;            col = t;    }  // row `edge`, all cols
            else if (t < 183) { row = 73 + (t - 128);  col = edge; }  // rows 73..127 at col edge
            else              { row = t - 183;         col = edge; }  // rows 0..54  at col edge
            val = ob[ch * HWSZ + row * WW + col];
        }
        s += val * val;
    }
    __shared__ float lds[8];
    int lane = threadIdx.x & 31, wid = threadIdx.x >> 5;
    s = waveSum32(s);
    if (lane == 0) lds[wid] = s;
    __syncthreads();
    if (threadIdx.x == 0) {
        float a = 0.f;
        #pragma unroll
        for (int w = 0; w < 8; ++w) a += lds[w];
        part[blockIdx.x] = a;
    }
}

// ---------------- kernel 3: interior PDE residuals ----------------

// torch.gradient 1st derivative at `pos` in [0,125]
__device__ __forceinline__ float g1(float fm, float f0, float fp, int pos, float inv_h) {
    if (pos == 0)   return (fp - f0) * inv_h;
    if (pos == 125) return (f0 - fm) * inv_h;
    return (fp - fm) * (0.5f * inv_h);
}

struct D12 { float d1, d2; };

// nested-gradient d1 & d2 along a direction with element stride `st`.
// `c` = interior coordinate (0..125), `idx` = full-grid linear index of center.
__device__ __forceinline__ D12 d12(const float* __restrict__ f, int idx, int st,
                                   int c, float inv_h) {
    float fm1 = f[idx - st], f0 = f[idx], fp1 = f[idx + st];
    // clamp the ±2 taps at full-grid bounds (values unused in those cases)
    int om2 = (c - 1 > 0)   ? 2 * st : st;   // c<=1 -> unused (clamped); c>=2 -> needs full -2
    int op2 = (c + 1 < 125) ? 2 * st : st;
    float fm2 = f[idx - om2];
    float fp2 = f[idx + op2];
    float gc = g1(fm1, f0, fp1, c,     inv_h);
    float gm = g1(fm2, fm1, f0, c - 1, inv_h);   // unused when c==0
    float gp = g1(f0, fp1, fp2, c + 1, inv_h);   // unused when c==125
    D12 d;
    d.d1 = gc;
    d.d2 = g1(gm, gc, gp, c, inv_h);
    return d;
}

__device__ __forceinline__ float d1only(const float* __restrict__ f, int idx, int st,
                                        int c, float inv_h) {
    return g1(f[idx - st], f[idx], f[idx + st], c, inv_h);
}

__global__ void k_int(const float* __restrict__ inp, const float* __restrict__ outp,
                      float* __restrict__ part_cont, float* __restrict__ part_ns) {
    const float inv_sx = 127.0f / 1.2f;
    const float inv_sy = 127.0f / 0.8f;
    float s_cont = 0.f, s_ns = 0.f;
    for (int i = blockIdx.x * blockDim.x + threadIdx.x; i < NTOT_INT;
         i += gridDim.x * blockDim.x) {
        int b = i / (NINT * NINT);
        int r = i - b * (NINT * NINT);
        int y = r / NINT;
        int x = r - y * NINT;
        int idx = (y + 1) * WW + (x + 1);          // full-grid center
        const float* pb = outp + b * CHW;          // p
        const float* ub = pb + HWSZ;               // u
        const float* vb = pb + 2 * HWSZ;           // v
        const float* ab = inp + b * HWSZ;          // alpha

        float u0 = ub[idx], v0 = vb[idx], alpha = ab[idx];

        D12 ux = d12(ub, idx, 1,  x, inv_sx);
        D12 uy = d12(ub, idx, WW, y, inv_sy);
        D12 vx = d12(vb, idx, 1,  x, inv_sx);
        D12 vy = d12(vb, idx, WW, y, inv_sy);
        float px = d1only(pb, idx, 1,  x, inv_sx);
        float py = d1only(pb, idx, WW, y, inv_sy);

        // RHO = MU = 1
        float xr = u0 * ux.d1 + v0 * uy.d1 + px - (ux.d2 + uy.d2) + alpha * u0;
        float yr = u0 * vx.d1 + v0 * vy.d1 + py - (vx.d2 + vy.d2) + alpha * v0;
        float c  = ux.d1 + vy.d1;
        s_cont += c * c;
        s_ns   += xr * xr + yr * yr;
    }
    __shared__ float lds[16];
    int lane = threadIdx.x & 31, wid = threadIdx.x >> 5;
    s_cont = waveSum32(s_cont);
    s_ns   = waveSum32(s_ns);
    if (lane == 0) { lds[wid] = s_cont; lds[8 + wid] = s_ns; }
    __syncthreads();
    if (threadIdx.x == 0) {
        float a = 0.f, bb = 0.f;
        #pragma unroll
        for (int w = 0; w < 8; ++w) { a += lds[w]; bb += lds[8 + w]; }
        part_cont[blockIdx.x] = a;
        part_ns[blockIdx.x]   = bb;
    }
}

// ---------------- kernel 4: finalize (WMMA reduction, 1 wave) ----------------

// Sum `n` floats (n % 64 == 0) with v_wmma_f32_16x16x4_f32.
// B = all-ones is invariant to the B lane layout; the full sum of A is
// invariant to the A layout permutation; D accumulates via C.
// sum(all D elements) = 16 * sum(A over all iterations).
__device__ __forceinline__ float wmma_sum(const float* __restrict__ p, int n, int lane) {
    v8f acc = {};
    v2f bones; bones.x = 1.0f; bones.y = 1.0f;
    for (int base = 0; base < n; base += 64) {
        v2f a;
        a.x = p[base + lane];
        a.y = p[base + 32 + lane];
        acc = __builtin_amdgcn_wmma_f32_16x16x4_f32(
            false, a, false, bones, (short)0, acc, false, false);
    }
    float t = acc[0] + acc[1] + acc[2] + acc[3] + acc[4] + acc[5] + acc[6] + acc[7];
    t = waveSum32(t);
    return t * (1.0f / 16.0f);
}

__global__ void k_final(const float* __restrict__ ws, float* __restrict__ out) {
    int lane = threadIdx.x;   // 32 threads, EXEC all ones (uniform control flow)
    float s_base = wmma_sum(ws,                          NB_BASE, lane);
    float s_cont = wmma_sum(ws + NB_BASE,                NB_INT,  lane);
    float s_ns   = wmma_sum(ws + NB_BASE + NB_INT,       NB_INT,  lane);
    float s_bc   = wmma_sum(ws + NB_BASE + 2 * NB_INT,   NB_BC,   lane);
    if (lane == 0) {
        const float n_int = (float)NTOT_INT;
        const float w_base = 0.1f / 300000.0f;                 // A_LOSS / SCALE_BASE
        const float w_cont = 0.3f / (n_int * 1e-5f);           // A_DIV / (N * SCALE_DIV)
        const float w_ns   = 0.3f / (2.0f * n_int * 0.01f);    // A_NS / (2N * SCALE_NS)
        const float w_bc   = 0.3f / (float)NTOT_BC;            // A_BC / N_bc
        out[0] = w_base * s_base + w_cont * s_cont + w_ns * s_ns + w_bc * s_bc;
    }
}

// ---------------- launcher ----------------

extern "C" void kernel_launch(void* const* d_in, const int* in_sizes, int n_in,
                              void* d_out, int out_size, void* d_ws, size_t ws_size,
                              hipStream_t stream) {
    const float* inputs  = (const float*)d_in[0];   // (256,1,128,128)
    const float* outputs = (const float*)d_in[1];   // (256,3,128,128)
    const float* targets = (const float*)d_in[2];   // (256,3,128,128)
    float* ws = (float*)d_ws;

    float* part_base = ws;                          // NB_BASE
    float* part_cont = ws + NB_BASE;                // NB_INT
    float* part_ns   = ws + NB_BASE + NB_INT;       // NB_INT
    float* part_bc   = ws + NB_BASE + 2 * NB_INT;   // NB_BC

    // every partial slot is written every call (grid size == array size),
    // so no zero-init pass is needed; all reductions are fixed-order.
    k_base<<<NB_BASE, 256, 0, stream>>>((const float4*)outputs, (const float4*)targets,
                                        part_base, (BATCH * CHW) / 4);
    k_bc  <<<NB_BC,   256, 0, stream>>>(outputs, part_bc);
    k_int <<<NB_INT,  256, 0, stream>>>(inputs, outputs, part_cont, part_ns);
    k_final<<<1, 32, 0, stream>>>(ws, (float*)d_out);
}